// ScaledDotProductAttention_66803921322282
// MI455X (gfx1250) — compile-verified
//
#include <hip/hip_runtime.h>

typedef __attribute__((ext_vector_type(2))) float v2f;
typedef __attribute__((ext_vector_type(8))) float v8f;

#define S_LEN   2048
#define D_HEAD  64
#define NHEADS  16
#define NBATCH  2

// One wave (32 lanes) owns a 16-query tile. 8 waves per block.
__global__ __launch_bounds__(256)
void attn_fp32_wmma_kernel(const float* __restrict__ Q,
                           const float* __restrict__ Kp,
                           const float* __restrict__ Vp,
                           const int*   __restrict__ mask,
                           float* __restrict__ out,
                           float* __restrict__ W)
{
    __shared__ float smax[8][16];
    __shared__ float sinv[8][16];

    const int tid  = threadIdx.x;
    const int wid  = tid >> 5;        // wave in block
    const int lane = tid & 31;
    const int lrow = lane & 15;       // lane within half-wave
    const int hlf  = lane >> 4;       // 0 or 1

    const int gw = blockIdx.x * 8 + wid;   // global 16-query tile id
    const int bh = gw >> 7;                // / (S/16 = 128)
    const int b  = bh >> 4;                // / H
    const int q0 = (gw & 127) << 4;

    const float* Qb = Q  + ((size_t)bh * S_LEN) * D_HEAD;
    const float* Kb = Kp + ((size_t)bh * S_LEN) * D_HEAD;
    const float* Vb = Vp + ((size_t)bh * S_LEN) * D_HEAD;
    float*       Wb = W  + ((size_t)bh * S_LEN) * S_LEN;
    const int*   Mb = mask + (size_t)b * S_LEN;

    // Invalidate read-only WGP-level cache lines (stale from previous graph replay).
    asm volatile("global_inv scope:SCOPE_DEV\n\t"
                 "s_wait_loadcnt 0x0" ::: "memory");

    // ---- Preload Q tile as 16 WMMA A-fragments (K-chunks of 4 along D) ----
    v2f aq[16];
    #pragma unroll
    for (int kc = 0; kc < 16; ++kc)
        aq[kc] = *(const v2f*)(Qb + (size_t)(q0 + lrow) * D_HEAD + kc * 4 + 2 * hlf);

    const float scale = 0.125f;   // 1/sqrt(64)
    const float log2e = 1.4426950408889634f;

    // ---- Pass A: raw scores -> W (weights region doubles as scratch); row max ----
    float mx[8];
    #pragma unroll
    for (int r = 0; r < 8; ++r) mx[r] = -1e30f;

    for (int kt = 0; kt < S_LEN / 16; ++kt) {
        const int k0 = kt * 16;
        v8f c = {0.f, 0.f, 0.f, 0.f, 0.f, 0.f, 0.f, 0.f};
        const float* kr = Kb + (size_t)(k0 + lrow) * D_HEAD + 2 * hlf;
        #pragma unroll
        for (int kc = 0; kc < 16; ++kc) {
            v2f bk = *(const v2f*)(kr + kc * 4);
            c = __builtin_amdgcn_wmma_f32_16x16x4_f32(false, aq[kc], false, bk,
                                                      (short)0, c, false, false);
        }
        const int mval = Mb[k0 + lrow];
        #pragma unroll
        for (int r = 0; r < 8; ++r) {
            float s = (mval == 0) ? -1e30f : c[r] * scale;
            mx[r] = fmaxf(mx[r], s);
            // C/D layout: row = r + 8*hlf, col = lrow
            Wb[(size_t)(q0 + r + 8 * hlf) * S_LEN + k0 + lrow] = s;
        }
    }

    // Row-max reduction across the 16 lanes of each half-wave.
    #pragma unroll
    for (int m = 1; m < 16; m <<= 1) {
        #pragma unroll
        for (int r = 0; r < 8; ++r)
            mx[r] = fmaxf(mx[r], __shfl_xor(mx[r], m, 32));
    }
    if (lrow == 0) {
        #pragma unroll
        for (int r = 0; r < 8; ++r) smax[wid][r + 8 * hlf] = mx[r];
    }
    // Raw-score stores must be visible to same-wave cross-lane reloads.
    asm volatile("s_wait_storecnt 0x0" ::: "memory");
    __syncthreads();

    // ---- Pass B: p = exp(s - m); row sums; P @ V via WMMA ----
    const float mrow = smax[wid][lrow];

    v8f acc[4];
    #pragma unroll
    for (int dc = 0; dc < 4; ++dc)
        acc[dc] = (v8f){0.f, 0.f, 0.f, 0.f, 0.f, 0.f, 0.f, 0.f};
    float sum = 0.f;

    const float* wr = Wb + (size_t)(q0 + lrow) * S_LEN + 2 * hlf;
    for (int kt = 0; kt < S_LEN / 16; ++kt) {
        const int k0 = kt * 16;
        v2f pa[4];
        #pragma unroll
        for (int ks = 0; ks < 4; ++ks) {
            // A-fragment layout read: row = lrow, cols = k0 + ks*4 + 2*hlf + {0,1}
            v2f s2 = *(const v2f*)(wr + k0 + ks * 4);
            v2f p;
            p.x = exp2f((s2.x - mrow) * log2e);
            p.y = exp2f((s2.y - mrow) * log2e);
            sum += p.x + p.y;
            pa[ks] = p;
        }
        #pragma unroll
        for (int ks = 0; ks < 4; ++ks) {
            // V B-fragment: b.x = V[k0+ks*4+2*hlf][dc*16+lrow], b.y = next key row
            const float* vr = Vb + (size_t)(k0 + ks * 4 + 2 * hlf) * D_HEAD + lrow;
            #pragma unroll
            for (int dc = 0; dc < 4; ++dc) {
                v2f bv;
                bv.x = vr[dc * 16];
                bv.y = vr[dc * 16 + D_HEAD];
                acc[dc] = __builtin_amdgcn_wmma_f32_16x16x4_f32(false, pa[ks], false, bv,
                                                                (short)0, acc[dc], false, false);
            }
        }
    }
    // Lane m and lane m+16 partition row lrow -> pairwise combine.
    sum += __shfl_xor(sum, 16, 32);
    if (hlf == 0) sinv[wid][lrow] = 1.0f / sum;
    __syncthreads();

    // ---- Write normalized output ----
    #pragma unroll
    for (int r = 0; r < 8; ++r) {
        const int row = r + 8 * hlf;
        const float iv = sinv[wid][row];
        float* orow = out + ((size_t)bh * S_LEN + q0 + row) * D_HEAD + lrow;
        #pragma unroll
        for (int dc = 0; dc < 4; ++dc)
            orow[dc * 16] = acc[dc][r] * iv;
    }

    // ---- Pass C: rewrite W as normalized probabilities (coalesced) ----
    for (int r = 0; r < 16; ++r) {
        const float mr = smax[wid][r];
        const float iv = sinv[wid][r];
        float* wrow = Wb + (size_t)(q0 + r) * S_LEN;
        for (int c0 = lane; c0 < S_LEN; c0 += 32) {
            float s = wrow[c0];
            wrow[c0] = exp2f((s - mr) * log2e) * iv;
        }
    }
}

extern "C" void kernel_launch(void* const* d_in, const int* in_sizes, int n_in,
                              void* d_out, int out_size, void* d_ws, size_t ws_size,
                              hipStream_t stream) {
    const float* Q = (const float*)d_in[0];
    const float* K = (const float*)d_in[1];
    const float* V = (const float*)d_in[2];
    const int*   M = (const int*)d_in[3];

    float* out = (float*)d_out;                                     // [B,H,S,D]
    float* W   = out + (size_t)NBATCH * NHEADS * S_LEN * D_HEAD;    // [B,H,S,S]

    const int total_tiles = NBATCH * NHEADS * (S_LEN / 16);         // 4096 waves
    const int blocks = total_tiles / 8;                             // 512 blocks x 256 thr

    attn_fp32_wmma_kernel<<<blocks, 256, 0, stream>>>(Q, K, V, M, out, W);
}